// AngularPenaltySMLoss_87986700026372
// MI455X (gfx1250) — compile-verified
//
#include <hip/hip_runtime.h>
#include <math.h>

#define BLOCK 256
#define ROWS_PER_BLOCK 16
#define CHUNK 4096   // elements per TDM tile (16 KB)
#define NBUF  4      // LDS ring buffers (prefetch distance 3)

typedef float        v4f __attribute__((ext_vector_type(4)));
typedef unsigned int v4u __attribute__((ext_vector_type(4)));
typedef int          v4i __attribute__((ext_vector_type(4)));
typedef int          v8i __attribute__((ext_vector_type(8)));

// ---- constants (S=32, M=0.5) ----
#define S_SCALE 32.0f
#define LOG2E   1.4426950408889634f
#define KS      46.166241308446827f   /* S * log2(e) */
#define LN2     0.6931471805599453f
#define COSM    0.8775825618903728f   /* cos(0.5) */
#define SINM    0.4794255386042030f   /* sin(0.5) */
#define EPS_C   1e-7f

__device__ __forceinline__ float fast_exp2(float x) {
#if __has_builtin(__builtin_amdgcn_exp2f)
  return __builtin_amdgcn_exp2f(x);
#else
  return exp2f(x);
#endif
}
__device__ __forceinline__ float fast_log2(float x) {
#if __has_builtin(__builtin_amdgcn_logf)
  return __builtin_amdgcn_logf(x);
#else
  return log2f(x);
#endif
}

#define HAVE_TDM (__has_builtin(__builtin_amdgcn_tensor_load_to_lds) || \
                  __has_builtin(__builtin_amdgcn_tensor_load_to_lds_d2))

#if HAVE_TDM
// Issue a 1-D TDM load of `nelem` f32 from global `gptr` to LDS byte offset
// `lds_off`. D# layout per CDNA5 ISA §8.3/8.4:
//   g0: count=1 | lds_addr | global_addr[56:0] | type=2
//   g1: data_size=2 (4B), tensor_dim0 = tile_dim0 = nelem, tensor_dim1 = 1,
//       tensor_dim0_stride = nelem, everything else 0 (1-D tile)
// cpol = 1 (TH_LOAD_NT): stream-once data, don't retain in L2.
__device__ __forceinline__ void tdm_load_1d(const float* gptr, unsigned lds_off,
                                            unsigned nelem) {
  unsigned long long ga = (unsigned long long)(size_t)gptr;
  v4u g0;
  g0[0] = 1u;                                            // count=1, user mode
  g0[1] = lds_off;                                       // lds_addr (bytes)
  g0[2] = (unsigned)(ga & 0xffffffffu);                  // global_addr[31:0]
  g0[3] = (unsigned)((ga >> 32) & 0x01ffffffu) | (2u << 30); // [56:32] | type=2
  v8i g1;
  g1[0] = (int)(2u << 16);                               // data_size = 4 bytes
  g1[1] = (int)((nelem & 0xffffu) << 16);                // tensor_dim0[15:0]
  g1[2] = (int)(((nelem >> 16) & 0xffffu) | (1u << 16)); // td0[31:16] | tensor_dim1=1
  g1[3] = (int)((nelem & 0xffffu) << 16);                // tile_dim0 = nelem
  g1[4] = 0;                                             // tile_dim1/2 unused
  g1[5] = (int)nelem;                                    // tensor_dim0_stride[31:0]
  g1[6] = 0;
  g1[7] = 0;
#if __has_builtin(__builtin_amdgcn_tensor_load_to_lds_d2)
  __builtin_amdgcn_tensor_load_to_lds_d2(g0, g1, 1);
#else
  v4i z4 = {0, 0, 0, 0};
  v8i z8 = {0, 0, 0, 0, 0, 0, 0, 0};
  // clang-23 / therock signature: (v4u g0, v8i g1, v4i g2, v4i g3, v8i, i32 cpol)
  __builtin_amdgcn_tensor_load_to_lds(g0, g1, z4, z4, z8, 1);
#endif
}

// Wait until at most `ahead` TDM descriptors remain outstanding (uniform arg,
// builtin wants an immediate -> small switch).
__device__ __forceinline__ void wait_tensorcnt_le(int ahead) {
#if __has_builtin(__builtin_amdgcn_s_wait_tensorcnt)
  switch (ahead) {
    case 0: __builtin_amdgcn_s_wait_tensorcnt(0); break;
    case 1: __builtin_amdgcn_s_wait_tensorcnt(1); break;
    case 2: __builtin_amdgcn_s_wait_tensorcnt(2); break;
    default: __builtin_amdgcn_s_wait_tensorcnt(3); break;
  }
#else
  switch (ahead) {
    case 0: asm volatile("s_wait_tensorcnt 0x0" ::: "memory"); break;
    case 1: asm volatile("s_wait_tensorcnt 0x1" ::: "memory"); break;
    case 2: asm volatile("s_wait_tensorcnt 0x2" ::: "memory"); break;
    default: asm volatile("s_wait_tensorcnt 0x3" ::: "memory"); break;
  }
#endif
}
#endif // HAVE_TDM

// ---------------------------------------------------------------------------
// Kernel 1: per-block ArcFace partial loss over ROWS_PER_BLOCK contiguous rows
// ---------------------------------------------------------------------------
__global__ void __launch_bounds__(BLOCK)
arcface_rows(const float* __restrict__ score, const int* __restrict__ labels,
             float* __restrict__ partials, int N, int C) {
  __shared__ __align__(16) float lds_buf[NBUF * CHUNK];
  __shared__ float wred[BLOCK / 32];

  const int tid  = threadIdx.x;
  const int row0 = blockIdx.x * ROWS_PER_BLOCK;
  const int rowEnd = (row0 + ROWS_PER_BLOCK < N) ? row0 + ROWS_PER_BLOCK : N;
  float Lsum = 0.0f;

#if HAVE_TDM
  const int chunksPerRow = C / CHUNK;
  const int totalChunks  = (rowEnd - row0) * chunksPerRow;
  const float* base = score + (size_t)row0 * (size_t)C;

  if (row0 < N) {
    if (tid < 32) {
      // prime the pipeline: chunks 0..2 -> buffers 0..2 (prefetch distance 3)
      for (int p = 0; p < NBUF - 1 && p < totalChunks; ++p)
        tdm_load_1d(base + (size_t)p * CHUNK,
                    (unsigned)(size_t)&lds_buf[p * CHUNK], CHUNK);
    }
    int j = 0;  // linear chunk index over this block's contiguous region
    for (int r = row0; r < rowEnd; ++r) {
      float acc  = 0.0f;   // per-thread partial of sum(exp2(KS*x)) for row r
      float tval = 0.0f;   // target-class cosine (valid in thread 0)
      const int lab = labels[r];
      for (int c = 0; c < chunksPerRow; ++c, ++j) {
        float* buf = &lds_buf[(j % NBUF) * CHUNK];
        if (tid < 32) {
          const int jpre = j + (NBUF - 1);
          if (jpre < totalChunks) {
            // buffer (jpre % NBUF) held chunk j-1: consumed before the
            // previous iteration's closing barrier -> safe to overwrite.
            tdm_load_1d(base + (size_t)jpre * CHUNK,
                        (unsigned)(size_t)&lds_buf[(jpre % NBUF) * CHUNK],
                        CHUNK);
          }
          // TDM completes in order: chunk j is done once at most
          // (last_issued - j) younger descriptors remain outstanding.
          int ahead = totalChunks - 1 - j;
          if (ahead > NBUF - 1) ahead = NBUF - 1;
          wait_tensorcnt_le(ahead);
        }
        __syncthreads();            // publish LDS data to all 8 waves
        const v4f* b4 = (const v4f*)buf;
        #pragma unroll 4
        for (int i = tid; i < CHUNK / 4; i += BLOCK) {
          v4f v = b4[i];
          acc += fast_exp2(v.x * KS);
          acc += fast_exp2(v.y * KS);
          acc += fast_exp2(v.z * KS);
          acc += fast_exp2(v.w * KS);
        }
        if (tid == 0) {
          const int lo = c * CHUNK;
          if (lab >= lo && lab < lo + CHUNK) tval = buf[lab - lo];
        }
        __syncthreads();            // all reads done before buf is re-filled
      }
      // wave32 reduction, then cross-wave via LDS
      #pragma unroll
      for (int off = 16; off > 0; off >>= 1) acc += __shfl_xor(acc, off, 32);
      if ((tid & 31) == 0) wred[tid >> 5] = acc;
      __syncthreads();
      if (tid == 0) {
        float rowsum = 0.0f;
        #pragma unroll
        for (int w = 0; w < BLOCK / 32; ++w) rowsum += wred[w];
        float t = fminf(fmaxf(tval, -1.0f + EPS_C), 1.0f - EPS_C);
        // cos(acos(t)+m) = t*cos(m) - sqrt(1-t^2)*sin(m)
        float numr  = S_SCALE * (t * COSM - sqrtf(fmaxf(1.0f - t * t, 0.0f)) * SINM);
        float excl  = rowsum - fast_exp2(t * KS);
        float denom = fast_exp2(numr * LOG2E) + excl;
        Lsum += numr - fast_log2(denom) * LN2;
      }
      __syncthreads();
    }
  }
#else
  // Fallback: direct global streaming (no TDM builtin available)
  for (int r = row0; r < rowEnd; ++r) {
    float acc = 0.0f;
    const v4f* g4 = (const v4f*)(score + (size_t)r * (size_t)C);
    for (int i = tid; i < C / 4; i += BLOCK) {
      v4f v = __builtin_nontemporal_load(&g4[i]);
      acc += fast_exp2(v.x * KS);
      acc += fast_exp2(v.y * KS);
      acc += fast_exp2(v.z * KS);
      acc += fast_exp2(v.w * KS);
    }
    #pragma unroll
    for (int off = 16; off > 0; off >>= 1) acc += __shfl_xor(acc, off, 32);
    if ((tid & 31) == 0) wred[tid >> 5] = acc;
    __syncthreads();
    if (tid == 0) {
      float rowsum = 0.0f;
      #pragma unroll
      for (int w = 0; w < BLOCK / 32; ++w) rowsum += wred[w];
      float tval = score[(size_t)r * (size_t)C + labels[r]];
      float t = fminf(fmaxf(tval, -1.0f + EPS_C), 1.0f - EPS_C);
      float numr  = S_SCALE * (t * COSM - sqrtf(fmaxf(1.0f - t * t, 0.0f)) * SINM);
      float excl  = rowsum - fast_exp2(t * KS);
      float denom = fast_exp2(numr * LOG2E) + excl;
      Lsum += numr - fast_log2(denom) * LN2;
    }
    __syncthreads();
  }
#endif
  if (tid == 0) partials[blockIdx.x] = Lsum;
}

// ---------------------------------------------------------------------------
// Kernel 2: deterministic reduction of block partials -> -sum/N
// ---------------------------------------------------------------------------
__global__ void __launch_bounds__(BLOCK)
arcface_finalize(const float* __restrict__ partials, int nblocks,
                 float* __restrict__ out, float invN) {
  __shared__ float s[BLOCK];
  float a = 0.0f;
  for (int i = threadIdx.x; i < nblocks; i += BLOCK) a += partials[i];
  s[threadIdx.x] = a;
  __syncthreads();
  #pragma unroll
  for (int off = BLOCK / 2; off > 0; off >>= 1) {
    if (threadIdx.x < off) s[threadIdx.x] += s[threadIdx.x + off];
    __syncthreads();
  }
  if (threadIdx.x == 0) out[0] = -s[0] * invN;
}

extern "C" void kernel_launch(void* const* d_in, const int* in_sizes, int n_in,
                              void* d_out, int out_size, void* d_ws, size_t ws_size,
                              hipStream_t stream) {
  const float* score  = (const float*)d_in[0];
  const int*   labels = (const int*)d_in[1];
  const int N = in_sizes[1];
  const int C = in_sizes[0] / N;
  const int nblocks = (N + ROWS_PER_BLOCK - 1) / ROWS_PER_BLOCK;
  float* partials = (float*)d_ws;
  arcface_rows<<<dim3(nblocks), dim3(BLOCK), 0, stream>>>(score, labels,
                                                          partials, N, C);
  arcface_finalize<<<dim3(1), dim3(BLOCK), 0, stream>>>(partials, nblocks,
                                                        (float*)d_out,
                                                        1.0f / (float)N);
}